// iterative_reco_model_28295244546160
// MI455X (gfx1250) — compile-verified
//
#include <hip/hip_runtime.h>
#include <math.h>

typedef __attribute__((ext_vector_type(2))) float v2f;
typedef __attribute__((ext_vector_type(8))) float v8f;

#define HH 512
#define WW 512
#define NA 512
#define ND 736
#define NS 736
#define DBLK (ND / 16)          // 46 detector blocks of 16
#define SINO_ELEMS (NA * ND)    // 376832
#define IMG_ELEMS (HH * WW)     // 262144

// ---------------------------------------------------------------------------
// Kernel 1: img = x + reco (into workspace), and copy reco into d_out tail.
// ---------------------------------------------------------------------------
__global__ void __launch_bounds__(256)
prep_kernel(const float* __restrict__ x, const float* __restrict__ reco,
            float* __restrict__ img, float* __restrict__ out_reco) {
  int i = blockIdx.x * blockDim.x + threadIdx.x;
  if (i < IMG_ELEMS) {
    float r = reco[i];
    img[i]      = x[i] + r;
    out_reco[i] = r;
  }
}

// ---------------------------------------------------------------------------
// Bilinear sample with the reference's per-tap zero-outside semantics.
// Four UNCONDITIONAL clamped loads via 32-bit indices off the uniform base
// (lowers to global_load saddr form: SGPR64 base + VGPR32 offset — no 64-bit
// vector address arithmetic). Validity is folded into the interpolation
// weights so no load can be sunk into a branch and EXEC never changes.
//   result = (t00*fx0m + t01*fx1m)*fy0m + (t10*fx0m + t11*fx1m)*fy1m
// which equals sum_ij t_ij * mask_xi * mask_yj * wxi * wyj  (reference form).
// ---------------------------------------------------------------------------
__device__ __forceinline__ float bilin(const float* __restrict__ img,
                                       float px, float py) {
  float x0f = floorf(px);
  float y0f = floorf(py);
  float fx = px - x0f;
  float fy = py - y0f;
  int xi = (int)x0f;
  int yi = (int)y0f;

  int xc0 = min(max(xi, 0), WW - 1);
  int xc1 = min(max(xi + 1, 0), WW - 1);
  int yc0 = min(max(yi, 0), HH - 1);
  int yc1 = min(max(yi + 1, 0), HH - 1);

  int ro0 = yc0 << 9;                 // WW == 512
  int ro1 = yc1 << 9;
  float t00 = img[ro0 + xc0];
  float t01 = img[ro0 + xc1];
  float t10 = img[ro1 + xc0];
  float t11 = img[ro1 + xc1];

  // single unsigned-range compare per tap coordinate
  bool bx0 = ((unsigned)xi       < (unsigned)WW);
  bool bx1 = ((unsigned)(xi + 1) < (unsigned)WW);
  bool by0 = ((unsigned)yi       < (unsigned)HH);
  bool by1 = ((unsigned)(yi + 1) < (unsigned)HH);

  float fx0m = bx0 ? (1.0f - fx) : 0.0f;
  float fx1m = bx1 ? fx          : 0.0f;
  float fy0m = by0 ? (1.0f - fy) : 0.0f;
  float fy1m = by1 ? fy          : 0.0f;

  float row0 = fmaf(t01, fx1m, t00 * fx0m);
  float row1 = fmaf(t11, fx1m, t10 * fx0m);
  return fmaf(row1, fy1m, row0 * fy0m);
}

// ---------------------------------------------------------------------------
// Kernel 2: forward projection. One wave32 per (angle, 16-detector block).
// A-matrix (16x4 f32): lane L<16 holds rows M=L at K=0,1; lane L>=16 holds
// M=L-16 at K=2,3. B = ones(4x16) -> C accumulates per-row sample sums.
// ---------------------------------------------------------------------------
__global__ void __launch_bounds__(256)
proj_kernel(const float* __restrict__ img, const float* __restrict__ angles,
            float* __restrict__ sino) {
  const int lane  = threadIdx.x & 31;
  const int wave  = blockIdx.x * (256 / 32) + (threadIdx.x >> 5);
  const int a     = wave / DBLK;   // angle index
  const int dblk  = wave % DBLK;   // detector block

  const float theta = angles[a];
  float si, co;
  __sincosf(theta, &si, &co);

  const int   m     = lane & 15;                       // detector row in block
  const float kbase = (float)((lane >> 4) << 1);       // 0 or 2
  const float s     = (float)(dblk * 16 + m) - 0.5f * (float)(ND - 1);
  const float cx    = 0.5f * (float)(WW - 1);
  const float cy    = 0.5f * (float)(HH - 1);
  const float tcen  = 0.5f * (float)(NS - 1);

  const float sx = fmaf(s, co, cx);
  const float sy = fmaf(s, si, cy);

  v8f acc = {};                       // 16x16 f32 accumulator (all cols equal)
  v2f bones;
  bones[0] = 1.0f;
  bones[1] = 1.0f;

  #pragma unroll 4
  for (int t0 = 0; t0 < NS; t0 += 4) {
    float ta = (float)t0 + kbase - tcen;   // this lane's first sample
    float tb = ta + 1.0f;                  // this lane's second sample

    float pxa = fmaf(-ta, si, sx);
    float pya = fmaf( ta, co, sy);
    float pxb = fmaf(-tb, si, sx);
    float pyb = fmaf( tb, co, sy);

    v2f amat;
    amat[0] = bilin(img, pxa, pya);
    amat[1] = bilin(img, pxb, pyb);

    // D = A(16x4) * ones(4x16) + C  => row sums accumulate in every column
    acc = __builtin_amdgcn_wmma_f32_16x16x4_f32(
        /*neg_a=*/false, amat, /*neg_b=*/false, bones,
        /*c_mod=*/(short)0, acc, /*reuse_a=*/false, /*reuse_b=*/false);
  }

  // C/D layout: lane n (0..15) VGPR v holds D[M=v, N=n]; lanes 16..31 hold
  // D[M=v+8, N=n-16]. Column 0 lives in lane 0 (rows 0..7) and lane 16
  // (rows 8..15). STEP == 1.0 so no scale.
  if ((lane & 15) == 0) {
    int dbase = dblk * 16 + ((lane >> 4) << 3);      // +0 or +8
    float4* out = (float4*)(sino + a * ND + dbase);  // 16B aligned (736%4==0)
    float4 lo = make_float4(acc[0], acc[1], acc[2], acc[3]);
    float4 hi = make_float4(acc[4], acc[5], acc[6], acc[7]);
    out[0] = lo;
    out[1] = hi;
  }
}

// ---------------------------------------------------------------------------
extern "C" void kernel_launch(void* const* d_in, const int* in_sizes, int n_in,
                              void* d_out, int out_size, void* d_ws, size_t ws_size,
                              hipStream_t stream) {
  const float* x      = (const float*)d_in[0];   // [1,512,512]
  const float* reco   = (const float*)d_in[1];   // [1,512,512]
  const float* angles = (const float*)d_in[2];   // [512]

  float* sino     = (float*)d_out;               // [1,512,736]
  float* out_reco = (float*)d_out + SINO_ELEMS;  // [1,512,512]
  float* img      = (float*)d_ws;                // 1 MB scratch: x + reco

  // img = x + reco; out_reco = reco
  prep_kernel<<<IMG_ELEMS / 256, 256, 0, stream>>>(x, reco, img, out_reco);

  // 512 angles * 46 det-blocks = 23552 waves; 8 waves/block -> 2944 blocks
  const int total_waves = NA * DBLK;
  proj_kernel<<<total_waves / 8, 256, 0, stream>>>(img, angles, sino);
}